// AddSLoss_38946763440539
// MI455X (gfx1250) — compile-verified
//
#include <hip/hip_runtime.h>
#include <math.h>

typedef __attribute__((ext_vector_type(2))) float v2f;
typedef __attribute__((ext_vector_type(8))) float v8f;

#define BS 32
#define NP 2048
#define NT (NP / 16) // 128 tiles of 16 points

// Workspace layout (floats):
//   tf   : BS*NP*3    transformed points (row major xyz)
//   tfB  : BS*NT*64   WMMA B operand: [-2x,-2y | -2z, 1.0] lane-striped
//   tgA  : BS*NT*64   WMMA A operand: [ x,  y  |  z,  r2 ] lane-striped
//   part : BS*NT      per-(batch,querytile) partial distance sums
#define TF_OFF   0
#define TFB_OFF  (TF_OFF + BS * NP * 3)
#define TGA_OFF  (TFB_OFF + BS * NT * 64)
#define PART_OFF (TGA_OFF + BS * NT * 64)

// ---------------------------------------------------------------------------
// Kernel 1: transform model points and pack WMMA operands.
// One thread per point (BS*NP threads).
// A/B lane layout for V_WMMA_F32_16X16X4_F32 (ISA 7.12.2):
//   lane l in 0..15 : VGPR0=K0, VGPR1=K1   for row/col l
//   lane l in 16..31: VGPR0=K2, VGPR1=K3   for row/col l-16
// K3 carries the r2/1.0 pair so the WMMA computes r2[m] - 2*dot(tf,tgt)
// with an inline-zero C operand (no C loads in the hot loop).
// ---------------------------------------------------------------------------
__global__ void __launch_bounds__(256) pack_kernel(
    const float* __restrict__ target, const float* __restrict__ mp,
    const float* __restrict__ H, float* __restrict__ ws) {
  int gid = blockIdx.x * blockDim.x + threadIdx.x; // 0 .. BS*NP-1
  int b = gid >> 11;   // / NP
  int n = gid & (NP - 1);

  float* tf  = ws + TF_OFF;
  float* tfB = ws + TFB_OFF;
  float* tgA = ws + TGA_OFF;

  const float* h = H + b * 16; // 4x4 row-major
  float mx = mp[gid * 3 + 0], my = mp[gid * 3 + 1], mz = mp[gid * 3 + 2];
  // tf[e] = sum_d mp[d]*H[e][d] + H[e][3]
  float tx = fmaf(h[0], mx, fmaf(h[1], my, fmaf(h[2], mz, h[3])));
  float ty = fmaf(h[4], mx, fmaf(h[5], my, fmaf(h[6], mz, h[7])));
  float tz = fmaf(h[8], mx, fmaf(h[9], my, fmaf(h[10], mz, h[11])));
  tf[gid * 3 + 0] = tx;
  tf[gid * 3 + 1] = ty;
  tf[gid * 3 + 2] = tz;

  int tile = n >> 4;
  int l = n & 15;
  // B operand: queries scaled by -2, K3 = 1.0
  float* bp = tfB + (((b * NT + tile) * 32 + l) * 2);
  *(v2f*)bp        = (v2f){-2.0f * tx, -2.0f * ty}; // lanes 0-15: K0,K1
  *(v2f*)(bp + 32) = (v2f){-2.0f * tz, 1.0f};       // lanes 16-31: K2,K3

  // A operand: targets, K3 = |target|^2
  float gx = target[gid * 3 + 0], gy = target[gid * 3 + 1],
        gz = target[gid * 3 + 2];
  float r2 = fmaf(gx, gx, fmaf(gy, gy, gz * gz));
  float* ap = tgA + (((b * NT + tile) * 32 + l) * 2);
  *(v2f*)ap        = (v2f){gx, gy};
  *(v2f*)(ap + 32) = (v2f){gz, r2};
}

// ---------------------------------------------------------------------------
// Kernel 2: one wave per (batch, query tile). 128 WMMA iterations over ref
// tiles (C = inline 0), per-lane argmin over 8 C rows, cross-half merge,
// then the final per-query distance with sym select and a wave-level sum.
// ---------------------------------------------------------------------------
__global__ void __launch_bounds__(256) nn_kernel(
    const float* __restrict__ target, const int* __restrict__ idx,
    float* __restrict__ ws) {
  const float* tf  = ws + TF_OFF;
  const float* tfB = ws + TFB_OFF;
  const float* tgA = ws + TGA_OFF;
  float* part      = ws + PART_OFF;

  int lane = threadIdx.x & 31;
  int wid = (blockIdx.x * blockDim.x + threadIdx.x) >> 5; // 0 .. BS*NT-1
  int b = wid >> 7;        // / NT
  int qt = wid & (NT - 1);

  // B operand is fixed for this wave's 16 queries.
  v2f bmat = *(const v2f*)(tfB + (((b * NT + qt) * 32 + lane) * 2));
  const float* aBase = tgA + (size_t)(b * NT) * 64;

  float bestv = 3.0e38f;
  int besti = 0;

#pragma unroll 4
  for (int rj = 0; rj < NT; ++rj) {
    v2f amat = *(const v2f*)(aBase + ((rj * 32 + lane) * 2));
    v8f c = {}; // inline-zero C: score = r2[M] - 2*dot(tf[N], target[M])
    c = __builtin_amdgcn_wmma_f32_16x16x4_f32(
        /*neg_a=*/false, amat, /*neg_b=*/false, bmat,
        /*c_mod=*/(short)0, c, /*reuse_a=*/false, /*reuse_b=*/false);

    int mbase = (rj << 4) + ((lane >> 4) << 3);
#pragma unroll
    for (int v = 0; v < 8; ++v) {
      bool lt = c[v] < bestv; // strict: keep earliest index on ties
      besti = lt ? (mbase + v) : besti;
      bestv = lt ? c[v] : bestv;
    }
  }

  // Merge the two half-wave M ranges: lanes l and l+16 hold the same query.
  float ov = __shfl_xor(bestv, 16, 32);
  int oi = __shfl_xor(besti, 16, 32);
  if (ov < bestv || (ov == bestv && oi < besti)) {
    bestv = ov;
    besti = oi;
  }

  // Lanes 0-15: final distance for query n = qt*16 + lane.
  float d = 0.0f;
  if (lane < 16) {
    int n = (qt << 4) + lane;
    const float* tp = tf + ((size_t)b * NP + n) * 3;
    // idx in [0,16); SYM_LIST = even values 0..14 -> is_sym == (idx even)
    bool sym = ((idx[b] & 1) == 0);
    int sel = sym ? besti : n;
    const float* gp = target + ((size_t)b * NP + sel) * 3;
    float dx = tp[0] - gp[0];
    float dy = tp[1] - gp[1];
    float dz = tp[2] - gp[2];
    d = sqrtf(fmaf(dx, dx, fmaf(dy, dy, dz * dz)));
  }

  // Wave sum (lanes >= 16 contribute 0).
  d += __shfl_xor(d, 1, 32);
  d += __shfl_xor(d, 2, 32);
  d += __shfl_xor(d, 4, 32);
  d += __shfl_xor(d, 8, 32);
  d += __shfl_xor(d, 16, 32);
  if (lane == 0) part[wid] = d;
}

// ---------------------------------------------------------------------------
// Kernel 3: reduce 128 partials per batch -> mean distance.
// ---------------------------------------------------------------------------
__global__ void __launch_bounds__(32) reduce_kernel(const float* __restrict__ ws_part,
                                                    float* __restrict__ out) {
  int b = blockIdx.x;
  int lane = threadIdx.x;
  float s = 0.0f;
  for (int k = lane; k < NT; k += 32) s += ws_part[b * NT + k];
  s += __shfl_xor(s, 1, 32);
  s += __shfl_xor(s, 2, 32);
  s += __shfl_xor(s, 4, 32);
  s += __shfl_xor(s, 8, 32);
  s += __shfl_xor(s, 16, 32);
  if (lane == 0) out[b] = s * (1.0f / (float)NP);
}

extern "C" void kernel_launch(void* const* d_in, const int* in_sizes, int n_in,
                              void* d_out, int out_size, void* d_ws,
                              size_t ws_size, hipStream_t stream) {
  const float* target = (const float*)d_in[0]; // [32,2048,3]
  const float* mp     = (const float*)d_in[1]; // [32,2048,3]
  const int*   idx    = (const int*)d_in[2];   // [32,1]
  const float* H      = (const float*)d_in[3]; // [32,4,4]
  float* out = (float*)d_out;                  // [32]
  float* ws = (float*)d_ws;

  // Kernel 1: BS*NP threads
  pack_kernel<<<(BS * NP) / 256, 256, 0, stream>>>(target, mp, H, ws);
  // Kernel 2: BS*NT waves = BS*NT*32 threads
  nn_kernel<<<(BS * NT * 32) / 256, 256, 0, stream>>>(target, idx, ws);
  // Kernel 3: one wave per batch
  reduce_kernel<<<BS, 32, 0, stream>>>(ws + PART_OFF, out);
}